// FLNO1d_30004641530292
// MI455X (gfx1250) — compile-verified
//
#include <hip/hip_runtime.h>
#include <hip/hip_bf16.h>
#include <math.h>

// FNO-1d forward for MI455X (gfx1250). All heavy math mapped to
// V_WMMA_F32_16X16X4_F32 (wave32). h stays L2-resident in [B][W][S] layout.

typedef __attribute__((ext_vector_type(2))) float v2f;
typedef __attribute__((ext_vector_type(8))) float v8f;

#define Bk 32
#define Sk 8192
#define Wk 64
#define Mk 16
#define Lk 4

__device__ __forceinline__ float gelu_exact(float x) {
    return 0.5f * x * (1.0f + erff(x * 0.70710678118654752f));
}

// A fragment (16x4) from row-major storage, leading dim ld.
// Layout (ISA 7.12.2): lanes 0-15 -> K0(v0)/K1(v1); lanes 16-31 -> K2/K3.
__device__ __forceinline__ v2f load_a(const float* __restrict__ p, int ld, int lane) {
    int r = lane & 15;
    int h = (lane >> 4) << 1;
    v2f a;
    a.x = p[r * ld + h];
    a.y = p[r * ld + h + 1];
    return a;
}
// B fragment (4x16) from row-major K x N storage, leading dim ld.
__device__ __forceinline__ v2f load_b(const float* __restrict__ p, int ld, int lane) {
    int r = lane & 15;
    int h = (lane >> 4) << 1;
    v2f b;
    b.x = p[h * ld + r];
    b.y = p[(h + 1) * ld + r];
    return b;
}

// ---------------------------------------------------------------- basis tables
// fb[k][s]: k<16 -> cos(2*pi*k*s/S), k>=16 -> -sin(2*pi*(k-16)*s/S)   (rfft)
// ib[k][s]: irfft rows with 1/S and conjugate-symmetry factor 2 folded in.
__global__ void fno_basis(float* __restrict__ fb, float* __restrict__ ib) {
    int idx = blockIdx.x * blockDim.x + threadIdx.x;
    if (idx >= 32 * Sk) return;
    int s = idx & (Sk - 1);
    int k = idx >> 13;
    int m = k & 15;
    int ph = (m * s) & (Sk - 1);                 // exact phase reduction
    float ang = (float)ph * (6.283185307179586f / (float)Sk);
    float c = cosf(ang), sn = sinf(ang);
    fb[idx] = (k < 16) ? c : -sn;
    float iv;
    if (k == 0)       iv = 1.0f / (float)Sk;
    else if (k < 16)  iv = (2.0f / (float)Sk) * c;
    else if (k == 16) iv = 0.0f;                 // irfft ignores imag of DC
    else              iv = -(2.0f / (float)Sk) * sn;
    ib[idx] = iv;
}

// ---------------------------------------------------------------- fc0 (K=2)
__global__ void fno_fc0(const float* __restrict__ x, const float* __restrict__ w0,
                        const float* __restrict__ b0, float* __restrict__ h) {
    int idx = blockIdx.x * blockDim.x + threadIdx.x;   // over B*W*S
    if (idx >= Bk * Wk * Sk) return;
    int s = idx & (Sk - 1);
    int w = (idx >> 13) & (Wk - 1);
    int b = idx >> 19;
    float xv = x[b * Sk + s];
    float tv = (float)s * (1.0f / (float)(Sk - 1));
    float v = xv * w0[w] + tv * w0[Wk + w] + b0[w];
    h[idx] = gelu_exact(v);                            // h layout [B][W][S]
}

// ---------------------------------------------------------------- forward DFT
// X[row=(b*W+i)][0..15]=Xr, [16..31]=Xi.  GEMM [2048 x 8192] x [8192 x 32].
// One block per 16-row tile; 8 waves K-split S, LDS cross-wave reduce.
// A fragment is reused by both column-tile WMMAs each K-step.
__global__ __launch_bounds__(256) void fno_dft(const float* __restrict__ hin,
                                               const float* __restrict__ fb,
                                               float* __restrict__ X) {
    __shared__ float redC[8 * 256];
    __shared__ float redS[8 * 256];
    const int lane = threadIdx.x & 31;
    const int wave = threadIdx.x >> 5;
    const int rt = blockIdx.x;                         // 0..127
    const float* arow = hin + (size_t)rt * 16 * Sk;

    v8f accC = {};
    v8f accS = {};
    const int k0 = wave * (Sk / 8);
    for (int k = k0; k < k0 + Sk / 8; k += 4) {
        v2f a  = load_a(arow + k, Sk, lane);
        // fb is stored N-major ([32][S]) => B^T fetch has the A pattern.
        v2f bc = load_a(fb + k, Sk, lane);             // cols 0..15 (cos)
        v2f bs = load_a(fb + 16 * Sk + k, Sk, lane);   // cols 16..31 (-sin)
        accC = __builtin_amdgcn_wmma_f32_16x16x4_f32(false, a, false, bc, (short)0, accC, false, false);
        accS = __builtin_amdgcn_wmma_f32_16x16x4_f32(false, a, false, bs, (short)0, accS, false, false);
    }
#pragma unroll
    for (int j = 0; j < 8; ++j) {
        redC[wave * 256 + j * 32 + lane] = accC[j];
        redS[wave * 256 + j * 32 + lane] = accS[j];
    }
    __syncthreads();
    int t = threadIdx.x;
    float sc = 0.f, ss = 0.f;
#pragma unroll
    for (int w = 0; w < 8; ++w) { sc += redC[w * 256 + t]; ss += redS[w * 256 + t]; }
    int j = t >> 5, l = t & 31;
    int row = rt * 16 + j + ((l >> 4) << 3);           // C layout: M=j / j+8
    int n = l & 15;
    X[row * 32 + n]      = sc;
    X[row * 32 + 16 + n] = ss;
}

// ---------------------------------------------------------------- mode mixing
// om2[(b*W+o)*32 + m] = Re, +16+m = Im of sum_i (Xr+iXi)(wr+iwi)
__global__ void fno_modemix(const float* __restrict__ X, const float* __restrict__ wr,
                            const float* __restrict__ wi, float* __restrict__ om2) {
    int idx = blockIdx.x * blockDim.x + threadIdx.x;   // B*W*M
    if (idx >= Bk * Wk * Mk) return;
    int m = idx & 15;
    int o = (idx >> 4) & 63;
    int b = idx >> 10;
    const float* xb = X + b * Wk * 32;
    float orr = 0.f, oii = 0.f;
    for (int i = 0; i < Wk; ++i) {
        float xr = xb[i * 32 + m];
        float xi = xb[i * 32 + 16 + m];
        float r  = wr[(i * Wk + o) * Mk + m];
        float w  = wi[(i * Wk + o) * Mk + m];
        orr += xr * r - xi * w;
        oii += xr * w + xi * r;
    }
    om2[(b * Wk + o) * 32 + m]      = orr;
    om2[(b * Wk + o) * 32 + 16 + m] = oii;
}

// ---------------------------------------------------------------- fused layer
// hout[b,o,s] = gelu( sum_k om2[b,o,k]*ib[k,s] + sum_i cw[o,i]*hin[b,i,s] + cb[o] )
// One wave owns one 16-wide s-tile and ALL four o-tiles: the shared B fragment
// (ib / h) is fetched once per K-step and consumed by 4 WMMAs (4x less L2
// traffic on the B side, 4x WMMA density per load).
__global__ __launch_bounds__(256) void fno_invconv(const float* __restrict__ hin,
                                                   const float* __restrict__ om2,
                                                   const float* __restrict__ ib,
                                                   const float* __restrict__ cw,
                                                   const float* __restrict__ cb,
                                                   float* __restrict__ hout) {
    const int lane = threadIdx.x & 31;
    const int wave = threadIdx.x >> 5;
    const int s0 = (blockIdx.x * 8 + wave) * 16;
    const int b  = blockIdx.z;

    v8f acc0 = {};
    v8f acc1 = {};
    v8f acc2 = {};
    v8f acc3 = {};

    // spectral part: K = 32 over om2 rows (A, L1-resident) x ib (B, shared)
    const float* aom = om2 + (size_t)b * Wk * 32;
#pragma unroll
    for (int k = 0; k < 32; k += 4) {
        v2f bb = load_b(ib + (size_t)k * Sk + s0, Sk, lane);
        v2f a0 = load_a(aom + 0 * 32 * 32 + k, 32, lane);   // o-tile 0: rows 0..15
        v2f a1 = load_a(aom + 16 * 32 + k, 32, lane);       // rows 16..31
        v2f a2 = load_a(aom + 32 * 32 + k, 32, lane);       // rows 32..47
        v2f a3 = load_a(aom + 48 * 32 + k, 32, lane);       // rows 48..63
        acc0 = __builtin_amdgcn_wmma_f32_16x16x4_f32(false, a0, false, bb, (short)0, acc0, false, false);
        acc1 = __builtin_amdgcn_wmma_f32_16x16x4_f32(false, a1, false, bb, (short)0, acc1, false, false);
        acc2 = __builtin_amdgcn_wmma_f32_16x16x4_f32(false, a2, false, bb, (short)0, acc2, false, false);
        acc3 = __builtin_amdgcn_wmma_f32_16x16x4_f32(false, a3, false, bb, (short)0, acc3, false, false);
    }
    // 1x1 conv part: K = 64 over conv_w rows (A, L1-resident) x h (B, shared)
    const float* hb = hin + (size_t)b * Wk * Sk;
#pragma unroll
    for (int k = 0; k < 64; k += 4) {
        v2f bb = load_b(hb + (size_t)k * Sk + s0, Sk, lane);
        v2f a0 = load_a(cw + 0 * 64 + k, 64, lane);
        v2f a1 = load_a(cw + 16 * 64 + k, 64, lane);
        v2f a2 = load_a(cw + 32 * 64 + k, 64, lane);
        v2f a3 = load_a(cw + 48 * 64 + k, 64, lane);
        acc0 = __builtin_amdgcn_wmma_f32_16x16x4_f32(false, a0, false, bb, (short)0, acc0, false, false);
        acc1 = __builtin_amdgcn_wmma_f32_16x16x4_f32(false, a1, false, bb, (short)0, acc1, false, false);
        acc2 = __builtin_amdgcn_wmma_f32_16x16x4_f32(false, a2, false, bb, (short)0, acc2, false, false);
        acc3 = __builtin_amdgcn_wmma_f32_16x16x4_f32(false, a3, false, bb, (short)0, acc3, false, false);
    }

    int half = lane >> 4, n = lane & 15;
    v8f* accs[4] = { &acc0, &acc1, &acc2, &acc3 };
#pragma unroll
    for (int ot = 0; ot < 4; ++ot) {
        v8f acc = *accs[ot];
#pragma unroll
        for (int j = 0; j < 8; ++j) {
            int row = ot * 16 + j + (half << 3);
            float v = acc[j] + cb[row];
            hout[((size_t)b * Wk + row) * Sk + s0 + n] = gelu_exact(v);
        }
    }
}

// ---------------------------------------------------------------- head: fc1+fc2
__global__ __launch_bounds__(256) void fno_head(const float* __restrict__ hin,
                                                const float* __restrict__ w1, const float* __restrict__ b1,
                                                const float* __restrict__ w2, const float* __restrict__ b2,
                                                float* __restrict__ out) {
    const int lane = threadIdx.x & 31;
    const int wave = threadIdx.x >> 5;
    int tile = blockIdx.x * 8 + wave;                  // over B * (S/16)
    int st = tile & (Sk / 16 - 1);
    int b  = tile / (Sk / 16);
    int s0 = st * 16;
    // A[r][k=i] = hin[(b*W+i)*S + s0+r]  (k-major => B-style fetch pattern)
    const float* ab = hin + (size_t)b * Wk * Sk + s0;

    int half = lane >> 4, n = lane & 15;
    v8f rowsum = {};
    for (int jt = 0; jt < 8; ++jt) {                   // 128 fc1 cols = 8 tiles
        v8f acc = {};
        int n0 = jt * 16;
#pragma unroll
        for (int k = 0; k < 64; k += 4) {
            v2f a  = load_b(ab + (size_t)k * Sk, Sk, lane);
            v2f bb = load_b(w1 + k * 128 + n0, 128, lane);
            acc = __builtin_amdgcn_wmma_f32_16x16x4_f32(false, a, false, bb, (short)0, acc, false, false);
        }
        float bj  = b1[n0 + n];
        float w2j = w2[n0 + n];
#pragma unroll
        for (int j = 0; j < 8; ++j)
            rowsum[j] += gelu_exact(acc[j] + bj) * w2j;   // fc2 fused
    }
#pragma unroll
    for (int j = 0; j < 8; ++j) {                      // reduce 16 lanes per half
        float v = rowsum[j];
        v += __shfl_xor(v, 1, 32);
        v += __shfl_xor(v, 2, 32);
        v += __shfl_xor(v, 4, 32);
        v += __shfl_xor(v, 8, 32);
        if (n == 0) {
            int row = j + (half << 3);
            out[(size_t)b * Sk + s0 + row] = v + b2[0];
        }
    }
}

extern "C" void kernel_launch(void* const* d_in, const int* in_sizes, int n_in,
                              void* d_out, int out_size, void* d_ws, size_t ws_size,
                              hipStream_t stream) {
    const float* x        = (const float*)d_in[0];
    const float* fc0_w    = (const float*)d_in[1];
    const float* fc0_b    = (const float*)d_in[2];
    const float* fconv_wr = (const float*)d_in[3];
    const float* fconv_wi = (const float*)d_in[4];
    const float* conv_w   = (const float*)d_in[5];
    const float* conv_b   = (const float*)d_in[6];
    const float* fc1_w    = (const float*)d_in[7];
    const float* fc1_b    = (const float*)d_in[8];
    const float* fc2_w    = (const float*)d_in[9];
    const float* fc2_b    = (const float*)d_in[10];
    float* out = (float*)d_out;

    char* ws = (char*)d_ws;
    const size_t HS = (size_t)Bk * Wk * Sk * sizeof(float);   // 64 MB
    float* h0  = (float*)(ws);                 // ping
    float* h1  = (float*)(ws + HS);            // pong
    float* fb  = (float*)(ws + 2 * HS);                         // 32*S f32
    float* ib  = (float*)(ws + 2 * HS + (size_t)32 * Sk * 4);
    float* X   = (float*)(ws + 2 * HS + (size_t)64 * Sk * 4);   // [B*W][32]
    float* om2 = X + (size_t)Bk * Wk * 32;                      // [B*W][32]

    fno_basis<<<(32 * Sk + 255) / 256, 256, 0, stream>>>(fb, ib);
    fno_fc0<<<(Bk * Wk * Sk + 255) / 256, 256, 0, stream>>>(x, fc0_w, fc0_b, h0);

    float* hin = h0;
    float* hout = h1;
    for (int l = 0; l < Lk; ++l) {
        fno_dft<<<dim3(Bk * Wk / 16), 256, 0, stream>>>(hin, fb, X);
        fno_modemix<<<(Bk * Wk * Mk + 255) / 256, 256, 0, stream>>>(
            X, fconv_wr + (size_t)l * Wk * Wk * Mk, fconv_wi + (size_t)l * Wk * Wk * Mk, om2);
        fno_invconv<<<dim3(Sk / 16 / 8, 1, Bk), 256, 0, stream>>>(
            hin, om2, ib, conv_w + (size_t)l * Wk * Wk, conv_b + (size_t)l * Wk, hout);
        float* t = hin; hin = hout; hout = t;
    }
    fno_head<<<dim3(Bk * (Sk / 16) / 8), 256, 0, stream>>>(hin, fc1_w, fc1_b, fc2_w, fc2_b, out);
}